// MoEAdapterLayer_46334107189261
// MI455X (gfx1250) — compile-verified
//
#include <hip/hip_runtime.h>
#include <hip/hip_bf16.h>
#include <math.h>

// ---------------------------------------------------------------------------
// MoE adapter layer for MI455X (gfx1250), wave32 + V_WMMA_F32_16X16X4_F32,
// with double-buffered async global->LDS weight staging (ASYNCcnt path).
// Sparsity exploit: only the top-2 gated experts contribute to the output.
// ---------------------------------------------------------------------------

#define B_DIM 32
#define N_TOK 1024
#define D_DIM 768
#define F_DIM 64
#define E_DIM 8

typedef float v2f __attribute__((ext_vector_type(2)));
typedef float v8f __attribute__((ext_vector_type(8)));
typedef int v4i __attribute__((ext_vector_type(4)));

#if __has_builtin(__builtin_amdgcn_global_load_async_to_lds_b128) && \
    __has_builtin(__builtin_amdgcn_s_wait_asynccnt)
#define HAS_ASYNC_LDS 1
typedef __attribute__((address_space(1))) v4i* gas_v4i;  // global int4*
typedef __attribute__((address_space(3))) v4i* las_v4i;  // LDS int4*
#endif

// Copy 16 bytes global -> LDS. Async (ASYNCcnt) path when available.
__device__ __forceinline__ void cp16_g2l(const float* gsrc, float* ldst) {
#ifdef HAS_ASYNC_LDS
  __builtin_amdgcn_global_load_async_to_lds_b128(
      (gas_v4i)(void*)gsrc,   // generic -> AS1 addrspacecast (+ type)
      (las_v4i)(void*)ldst,   // generic -> AS3 addrspacecast (+ type)
      /*offset=*/0, /*cpol=*/0);
#else
  float4 v = *(const float4*)gsrc;
  *(float4*)ldst = v;
#endif
}

__device__ __forceinline__ void wait_stage_and_barrier() {
#ifdef HAS_ASYNC_LDS
  __builtin_amdgcn_s_wait_asynccnt(0);  // own async copies landed in LDS
#endif
  __syncthreads();                      // publish across the workgroup
}

__device__ __forceinline__ v8f wmma_f32_4(v2f a, v2f b, v8f c) {
  // D = A(16x4) * B(4x16) + C(16x16), all fp32, wave32.
  return __builtin_amdgcn_wmma_f32_16x16x4_f32(
      /*neg_a=*/false, a, /*neg_b=*/false, b,
      /*c_mod=*/(short)0, c, /*reuse_a=*/false, /*reuse_b=*/false);
}

__device__ __forceinline__ float gelu_tanh(float x) {
  // jax.nn.gelu default (tanh approximation)
  const float k0 = 0.7978845608028654f;  // sqrt(2/pi)
  const float k1 = 0.044715f;
  float x3 = x * x * x;
  float t = tanhf(k0 * (x + k1 * x3));
  return 0.5f * x * (1.0f + t);
}

// ---------------------------------------------------------------------------
// Kernel 1: partial mean-pool over N.  grid=(B, 8), block=256.
// ---------------------------------------------------------------------------
__global__ __launch_bounds__(256) void moe_pool_kernel(
    const float* __restrict__ tokens, float* __restrict__ partial) {
  const int b = blockIdx.x;
  const int s = blockIdx.y;
  const int t = threadIdx.x;
  const float* src = tokens + ((size_t)b * N_TOK + (size_t)s * 128) * D_DIM;
  float a0 = 0.f, a1 = 0.f, a2 = 0.f;
  for (int n = 0; n < 128; ++n) {
    const float* r = src + (size_t)n * D_DIM;
    a0 += r[t];
    a1 += r[t + 256];
    a2 += r[t + 512];
  }
  float* dst = partial + (size_t)(b * 8 + s) * D_DIM;
  dst[t] = a0;
  dst[t + 256] = a1;
  dst[t + 512] = a2;
}

// ---------------------------------------------------------------------------
// Kernel 2: gating.  1 block x 256 threads; thread = (b, e).
// ---------------------------------------------------------------------------
__global__ __launch_bounds__(256) void moe_gate_kernel(
    const float* __restrict__ w_gate, const float* __restrict__ partial,
    int* __restrict__ topIdx, float* __restrict__ topG,
    float* __restrict__ outTail) {
  __shared__ float logitsS[B_DIM * E_DIM];
  __shared__ float gatesS[B_DIM * E_DIM];
  const int tid = threadIdx.x;
  const int b = tid >> 3;
  const int e = tid & 7;

  float acc = 0.f;
  const float* pb = partial + (size_t)b * 8 * D_DIM;
  for (int d = 0; d < D_DIM; ++d) {
    float p = 0.f;
#pragma unroll
    for (int s = 0; s < 8; ++s) p += pb[s * D_DIM + d];
    acc += p * w_gate[d * E_DIM + e];
  }
  logitsS[tid] = acc * (1.0f / (float)N_TOK);
  __syncthreads();

  if (tid < B_DIM) {
    const float* L = &logitsS[tid * E_DIM];
    int i0 = 0;
    float v0 = L[0];
    for (int k = 1; k < E_DIM; ++k)
      if (L[k] > v0) { v0 = L[k]; i0 = k; }  // first-index tiebreak, as top_k
    int i1 = -1;
    float v1 = -3.4e38f;
    for (int k = 0; k < E_DIM; ++k)
      if (k != i0 && L[k] > v1) { v1 = L[k]; i1 = k; }
    float t = expf(v1 - v0);
    float inv = 1.0f / (1.0f + t);
    float g0 = inv;
    float g1 = t * inv;
#pragma unroll
    for (int k = 0; k < E_DIM; ++k) gatesS[tid * E_DIM + k] = 0.f;
    gatesS[tid * E_DIM + i0] = g0;
    gatesS[tid * E_DIM + i1] = g1;
    topIdx[tid * 2 + 0] = i0;
    topIdx[tid * 2 + 1] = i1;
    topG[tid * 2 + 0] = g0;
    topG[tid * 2 + 1] = g1;
  }
  __syncthreads();

  if (tid < E_DIM) {
    float imp = 0.f, ld = 0.f;
    for (int bb = 0; bb < B_DIM; ++bb) {
      float g = gatesS[bb * E_DIM + tid];
      imp += g;
      ld += (g > 0.f) ? 1.f : 0.f;
    }
    outTail[tid] = imp;         // importance[e]
    outTail[E_DIM + tid] = ld;  // load[e]
  }
}

// ---------------------------------------------------------------------------
// Kernel 3: expert MLP for the two gated experts, fp32 WMMA + async LDS
// weight staging.  grid=(B, N_TOK/64), block=128 (4 wave32s).
//   wstage: [2 buffers][2 experts][64][64] f32, 64KB, double-buffered,
//           reused (union) for the down K-chunks and the up D-chunks.
// ---------------------------------------------------------------------------
#define HPAD 66      // h row padding: conflict-free ds_load_b64 A-frags
#define CHUNK 64     // K-chunk (down) / D-chunk (up)
#define BUFW 8192    // floats per buffer (2 experts * 64 * 64)

// Stage down-chunk: w_down[e][k0..k0+63][0..63]  (contiguous 16KB per expert)
__device__ __forceinline__ void stage_down(float* wstage, int buf, int k0,
                                           const float* Wd0, const float* Wd1,
                                           int tid) {
  float* dstB = wstage + buf * BUFW;
#pragma unroll
  for (int e = 0; e < 2; ++e) {
    const float* src = (e ? Wd1 : Wd0) + k0 * F_DIM;
    float* dst = dstB + e * (CHUNK * F_DIM);
#pragma unroll
    for (int r = 0; r < 8; ++r) {
      const int p = r * 128 + tid;  // 16B piece id (1024 pieces / expert)
      cp16_g2l(src + p * 4, dst + p * 4);
    }
  }
}

// Stage up-chunk: w_up[e][0..63][d0..d0+63]  (64 strided rows of 256B)
__device__ __forceinline__ void stage_up(float* wstage, int buf, int d0,
                                         const float* Wu0, const float* Wu1,
                                         int tid) {
  float* dstB = wstage + buf * BUFW;
#pragma unroll
  for (int e = 0; e < 2; ++e) {
    const float* src = (e ? Wu1 : Wu0) + d0;
    float* dst = dstB + e * (F_DIM * CHUNK);
#pragma unroll
    for (int r = 0; r < 8; ++r) {
      const int p = r * 128 + tid;
      const int row = p >> 4;        // 0..63 (k index)
      const int c4 = (p & 15) * 4;   // float offset within 256B row
      cp16_g2l(src + row * D_DIM + c4, dst + row * CHUNK + c4);
    }
  }
}

__global__ __launch_bounds__(128) void moe_expert_kernel(
    const float* __restrict__ tokens, const float* __restrict__ w_down,
    const float* __restrict__ b_down, const float* __restrict__ w_up,
    const float* __restrict__ b_up, const int* __restrict__ topIdx,
    const float* __restrict__ topG, float* __restrict__ out) {
  const int b = blockIdx.x;
  const int tid = threadIdx.x;
  const int wave = tid >> 5;
  const int lane = tid & 31;
  const int lo = lane & 15;  // M row (A) / N col (B,C,D)
  const int hi = lane >> 4;  // K-half select (A,B) / M-half select (C,D)
  const int nBase = blockIdx.y * 64 + wave * 16;

  const int e0 = topIdx[b * 2 + 0];
  const int e1 = topIdx[b * 2 + 1];
  const float g0 = topG[b * 2 + 0];
  const float g1 = topG[b * 2 + 1];
  const float gsum = g0 + g1;

  __shared__ float ldsH[4 * 2 * 16 * HPAD];   // 33.8 KB: gelu(h) per wave
  __shared__ float wstage[2 * BUFW];          // 64 KB: weight double-buffer
  float* hBase = &ldsH[wave * (2 * 16 * HPAD)];

  const float* Arow = tokens + ((size_t)b * N_TOK + nBase + lo) * D_DIM;
  const float* Wd0 = w_down + (size_t)e0 * D_DIM * F_DIM;
  const float* Wd1 = w_down + (size_t)e1 * D_DIM * F_DIM;

  // ======== down projection: h[16x64] for both experts ====================
  v8f accD[2][4];
#pragma unroll
  for (int e = 0; e < 2; ++e)
#pragma unroll
    for (int f = 0; f < 4; ++f) accD[e][f] = (v8f)0.0f;

  stage_down(wstage, 0, 0, Wd0, Wd1, tid);  // prime buffer 0

  for (int c = 0; c < D_DIM / CHUNK; ++c) {
    const int cur = c & 1;
    wait_stage_and_barrier();  // chunk c resident; all waves done with prev
    if (c + 1 < D_DIM / CHUNK)
      stage_down(wstage, 1 - cur, (c + 1) * CHUNK, Wd0, Wd1, tid);

    const float* Bc = wstage + cur * BUFW;
    const int kBase = c * CHUNK;
#pragma unroll 4
    for (int kk = 0; kk < CHUNK; kk += 4) {
      // A fragment: lane(lo,hi) holds tokens[nBase+lo][kBase+kk+2*hi+{0,1}]
      v2f a = *(const v2f*)(Arow + kBase + kk + 2 * hi);
      __builtin_prefetch(Arow + kBase + kk + 96, 0, 3);
      const int rk = (kk + 2 * hi) * F_DIM;
#pragma unroll
      for (int f = 0; f < 4; ++f) {
        const int col = f * 16 + lo;
        v2f b0;
        b0.x = Bc[rk + col];
        b0.y = Bc[rk + F_DIM + col];
        accD[0][f] = wmma_f32_4(a, b0, accD[0][f]);
        v2f b1;
        b1.x = Bc[CHUNK * F_DIM + rk + col];
        b1.y = Bc[CHUNK * F_DIM + rk + F_DIM + col];
        accD[1][f] = wmma_f32_4(a, b1, accD[1][f]);
      }
    }
  }
  __syncthreads();  // all waves done reading wstage before up staging reuses it

  // ======== prime up-stage, then bias + GELU -> LDS h =====================
  const float* Wu0 = w_up + (size_t)e0 * F_DIM * D_DIM;
  const float* Wu1 = w_up + (size_t)e1 * F_DIM * D_DIM;
  stage_up(wstage, 0, 0, Wu0, Wu1, tid);  // overlap with GELU below

#pragma unroll
  for (int e = 0; e < 2; ++e) {
    const int eIdx = e ? e1 : e0;
#pragma unroll
    for (int f = 0; f < 4; ++f) {
      const float bd = b_down[eIdx * F_DIM + f * 16 + lo];
#pragma unroll
      for (int v = 0; v < 8; ++v) {
        const int m = v + 8 * hi;  // C/D layout: vgpr v, half hi -> row m
        float h = gelu_tanh(accD[e][f][v] + bd);
        hBase[(e * 16 + m) * HPAD + f * 16 + lo] = h;
      }
    }
  }
  // h region is wave-private; same-wave DS ordering suffices.

  // ======== up projection + fused gate/bias/residual ======================
  for (int c = 0; c < D_DIM / CHUNK; ++c) {
    const int cur = c & 1;
    wait_stage_and_barrier();
    if (c + 1 < D_DIM / CHUNK)
      stage_up(wstage, 1 - cur, (c + 1) * CHUNK, Wu0, Wu1, tid);

    const float* Uc = wstage + cur * BUFW;
#pragma unroll
    for (int dt = 0; dt < CHUNK / 16; ++dt) {
      const int d0 = c * CHUNK + dt * 16;
      v8f acc0 = (v8f)0.0f;
      v8f acc1 = (v8f)0.0f;
#pragma unroll
      for (int ku = 0; ku < F_DIM; ku += 4) {
        const int kc = ku + 2 * hi;
        v2f ha0 = *(const v2f*)&hBase[(0 * 16 + lo) * HPAD + kc];
        v2f wb0;
        wb0.x = Uc[kc * CHUNK + dt * 16 + lo];
        wb0.y = Uc[(kc + 1) * CHUNK + dt * 16 + lo];
        acc0 = wmma_f32_4(ha0, wb0, acc0);
        v2f ha1 = *(const v2f*)&hBase[(1 * 16 + lo) * HPAD + kc];
        v2f wb1;
        wb1.x = Uc[F_DIM * CHUNK + kc * CHUNK + dt * 16 + lo];
        wb1.y = Uc[F_DIM * CHUNK + (kc + 1) * CHUNK + dt * 16 + lo];
        acc1 = wmma_f32_4(ha1, wb1, acc1);
      }
      const float bu0 = b_up[e0 * D_DIM + d0 + lo];
      const float bu1 = b_up[e1 * D_DIM + d0 + lo];
      const float biasTerm = g0 * bu0 + g1 * bu1;
#pragma unroll
      for (int v = 0; v < 8; ++v) {
        const int row = nBase + v + 8 * hi;
        const size_t off = ((size_t)b * N_TOK + row) * D_DIM + d0 + lo;
        const float x = tokens[off];  // residual (mostly L2-resident)
        out[off] = g0 * acc0[v] + g1 * acc1[v] + biasTerm + gsum * x;
      }
    }
  }
}

// ---------------------------------------------------------------------------
// Host launcher
// ---------------------------------------------------------------------------
extern "C" void kernel_launch(void* const* d_in, const int* in_sizes, int n_in,
                              void* d_out, int out_size, void* d_ws,
                              size_t ws_size, hipStream_t stream) {
  const float* tokens = (const float*)d_in[0];  // (32,1024,768)
  const float* w_gate = (const float*)d_in[1];  // (768,8)
  const float* w_down = (const float*)d_in[2];  // (8,768,64)
  const float* b_down = (const float*)d_in[3];  // (8,64)
  const float* w_up   = (const float*)d_in[4];  // (8,64,768)
  const float* b_up   = (const float*)d_in[5];  // (8,768)
  float* out = (float*)d_out;  // combined ++ importance(8) ++ load(8)

  const size_t partialBytes = (size_t)B_DIM * 8 * D_DIM * sizeof(float);
  float* partial = (float*)d_ws;
  int* topIdx = (int*)((char*)d_ws + partialBytes);
  float* topG = (float*)((char*)d_ws + partialBytes + 256);

  float* outTail = out + (size_t)B_DIM * N_TOK * D_DIM;

  moe_pool_kernel<<<dim3(B_DIM, 8), 256, 0, stream>>>(tokens, partial);
  moe_gate_kernel<<<1, 256, 0, stream>>>(w_gate, partial, topIdx, topG, outTail);
  moe_expert_kernel<<<dim3(B_DIM, N_TOK / 64), 128, 0, stream>>>(
      tokens, w_down, b_down, w_up, b_up, topIdx, topG, out);
}